// CubicalLayer_47828755808437
// MI455X (gfx1250) — compile-verified
//
#include <hip/hip_runtime.h>
#include <stdint.h>

// ---------------------------------------------------------------------------
// Persistence-diagram gather for a cubical complex (MI455X / gfx1250).
//   dgm[p] = (Xflat[idx[2p]], Xflat[idx[2p+1]]), zeroed when |d-b| <= 0.
// Memory-latency-bound random gather; index stream is staged into LDS with
// the CDNA5 async-to-LDS engine (ASYNCcnt), double buffered so index fetch
// overlaps the random gathers. Outputs use non-temporal stores.
// ---------------------------------------------------------------------------

#define TILE 256  // pairs per block-tile (== blockDim.x)

typedef float vf2 __attribute__((ext_vector_type(2)));

#if defined(__gfx1250__)
#define CDNA5_ASYNC 1
#else
#define CDNA5_ASYNC 0
#endif

#if CDNA5_ASYNC
// Per-lane async copy of 8 bytes (one int32 index pair) global -> LDS.
// VDST operand = VGPR holding the LDS byte offset, VADDR = 64-bit address.
__device__ __forceinline__ void async_ld_pair(unsigned lds_byte_off, const void* gaddr) {
    asm volatile("global_load_async_to_lds_b64 %0, %1, off"
                 :: "v"(lds_byte_off), "v"(gaddr)
                 : "memory");
}
__device__ __forceinline__ void wait_async_le1() {
    asm volatile("s_wait_asynccnt 1" ::: "memory");
}
#endif

__global__ void __launch_bounds__(TILE, 1)
diagram_gather(const float* __restrict__ X,
               const int*  __restrict__ idx,   // 2*npairs int32
               float*      __restrict__ out,   // 2*npairs float32
               int npairs)
{
    const int tid    = threadIdx.x;
    const int ntiles = (npairs + TILE - 1) / TILE;
    int t = blockIdx.x;
    if (t >= ntiles) return;

    const int2* gp = (const int2*)idx;

#if CDNA5_ASYNC
    __shared__ int2 sbuf[2][TILE];
    const unsigned lds_off0 = (unsigned)(uintptr_t)&sbuf[0][tid];
    const unsigned lds_off1 = (unsigned)(uintptr_t)&sbuf[1][tid];
    const int last = ntiles - 1;

    // Clamped per-lane source address for tile tt (keeps EXEC uniform on tail).
    auto src = [&](int tt) -> const void* {
        int p = tt * TILE + tid;
        if (p >= npairs) p = npairs - 1;
        return (const void*)(gp + p);
    };

    // Prologue: fill buffer 0 for the first tile.
    async_ld_pair(lds_off0, src(t));

    int buf = 0;
    for (; t < ntiles; t += (int)gridDim.x) {
        // Prefetch the next tile into the other buffer (clamped => redundant
        // but harmless on the final iteration; that buffer is never read).
        int tn = t + (int)gridDim.x;
        if (tn > last) tn = last;
        async_ld_pair(buf ? lds_off0 : lds_off1, src(tn));

        // Async ops retire in order: <=1 outstanding means the current
        // buffer's data has landed in LDS, while the prefetch stays in flight.
        wait_async_le1();

        int2 bi = sbuf[buf][tid];
        int p = t * TILE + tid;
        if (p < npairs) {
            float b = X[bi.x];          // random 4B gathers (L2 RT caching:
            float d = X[bi.y];          // ~2 touches per 128B line on average)
            bool keep = __builtin_fabsf(d - b) > 0.0f;  // NaN -> false -> zero
            vf2 o;
            o.x = keep ? b : 0.0f;
            o.y = keep ? d : 0.0f;
            // Streaming result, never re-read: bypass cache with NT store.
            __builtin_nontemporal_store(o, (vf2*)(out + 2 * (long)p));
        }
        buf ^= 1;
    }
#else
    for (; t < ntiles; t += (int)gridDim.x) {
        int p = t * TILE + tid;
        if (p < npairs) {
            int2 bi = gp[p];
            float b = X[bi.x];
            float d = X[bi.y];
            bool keep = __builtin_fabsf(d - b) > 0.0f;
            vf2 o;
            o.x = keep ? b : 0.0f;
            o.y = keep ? d : 0.0f;
            __builtin_nontemporal_store(o, (vf2*)(out + 2 * (long)p));
        }
    }
#endif
}

extern "C" void kernel_launch(void* const* d_in, const int* in_sizes, int n_in,
                              void* d_out, int out_size, void* d_ws, size_t ws_size,
                              hipStream_t stream) {
    const float* X    = (const float*)d_in[0];
    const int*   idx0 = (const int*)d_in[1];
    const int*   idx1 = (const int*)d_in[2];
    float*       out  = (float*)d_out;

    const int n0 = in_sizes[1] / 2;  // pairs in dim 0
    const int n1 = in_sizes[2] / 2;  // pairs in dim 1

    // ~2048 blocks => ~4 tiles/block: enough loop trips for the double-buffered
    // async pipeline, and ~16K waves x 64 outstanding gathers each to saturate
    // 23.3 TB/s at ~1us miss latency (Little's law needs ~150K lines in flight).
    const int MAXB = 2048;

    if (n0 > 0) {
        int tiles = (n0 + TILE - 1) / TILE;
        int blocks = tiles < MAXB ? tiles : MAXB;
        diagram_gather<<<blocks, TILE, 0, stream>>>(X, idx0, out, n0);
    }
    if (n1 > 0) {
        int tiles = (n1 + TILE - 1) / TILE;
        int blocks = tiles < MAXB ? tiles : MAXB;
        diagram_gather<<<blocks, TILE, 0, stream>>>(X, idx1, out + 2 * (long)n0, n1);
    }
}